// WaveletFeatureExtractor_83803401880523
// MI455X (gfx1250) — compile-verified
//
#include <hip/hip_runtime.h>
#include <hip/hip_bf16.h>

typedef __attribute__((ext_vector_type(2))) float v2f;
typedef __attribute__((ext_vector_type(8))) float v8f;

// db4 decomposition filters, pre-reversed for the reference's true-convolution:
// out[j] = sum_i REV[i] * x_sym(2j - 6 + i),  REV[i] = DEC[7-i]
__device__ __constant__ float REV_LO[8] = {
     0.2303778133088965f,   0.7148465705529157f,   0.6308807679298589f,
    -0.027983769416859854f, -0.18703481171909309f, 0.030841381835560764f,
     0.0328830116668852f,  -0.010597401785069032f };
__device__ __constant__ float REV_HI[8] = {
    -0.010597401785069032f, -0.0328830116668852f,  0.030841381835560764f,
     0.18703481171909309f,  -0.027983769416859854f, -0.6308807679298589f,
     0.7148465705529157f,   -0.2303778133088965f };

__device__ __forceinline__ int sym_idx(int t, int n) {
    return t < 0 ? (-1 - t) : (t >= n ? (2 * n - 1 - t) : t);
}

// Low-pass DWT level: in[n] -> outA[m], m = floor((n+7)/2). 256 threads.
__device__ __forceinline__ void dwt_lo(const float* __restrict__ in, int n,
                                       float* __restrict__ outA) {
    const int m = (n + 7) >> 1;
    for (int j = threadIdx.x; j < m; j += 256) {
        float a = 0.f;
        const int tb = 2 * j - 6;
#pragma unroll
        for (int i = 0; i < 8; ++i)
            a += REV_LO[i] * in[sym_idx(tb + i, n)];
        outA[j] = a;
    }
}

// Adaptive-pool(128) of the HIGH-pass band of dwt(in[n]) computed on the fly
// (never materialized). Threads 0..127 each own one bin.
__device__ __forceinline__ void pool_hi_dwt(const float* __restrict__ in, int n,
                                            float* __restrict__ gout) {
    const int m = (n + 7) >> 1;
    const int p = threadIdx.x;
    if (p < 128) {
        const int s = (p * m) >> 7;
        const int e = ((p + 1) * m + 127) >> 7;
        float sum = 0.f;
        for (int j = s; j < e; ++j) {
            float d = 0.f;
            const int tb = 2 * j - 6;
#pragma unroll
            for (int i = 0; i < 8; ++i)
                d += REV_HI[i] * in[sym_idx(tb + i, n)];
            sum += d;
        }
        gout[p] = sum / (float)(e - s);
    }
}

__device__ __forceinline__ void pool_plain(const float* __restrict__ in, int n,
                                           float* __restrict__ gout) {
    const int p = threadIdx.x;
    if (p < 128) {
        const int s = (p * n) >> 7;
        const int e = ((p + 1) * n + 127) >> 7;
        float sum = 0.f;
        for (int j = s; j < e; ++j) sum += in[j];
        gout[p] = sum / (float)(e - s);
    }
}

// Kernel 1: one block per batch row. cA cascade lives in LDS; cD bands are
// pooled on the fly from the parent cA (cD1 is unused by the reference).
__global__ __launch_bounds__(256) void dwt_pool_kernel(
    const float* __restrict__ x,       // [2048, 16384]
    float* __restrict__ pooled) {      // [2048, 5, 128]
    __shared__ float bufA[8196];       // cA1 / cA3 / cA5
    __shared__ float bufB[4104];       // cA2 / cA4
    const int b = blockIdx.x;
    const float* xr = x + (size_t)b * 16384;
    float* pout = pooled + (size_t)b * 640;

    dwt_lo(xr, 16384, bufA);                 // cA1, n=8195
    __syncthreads();
    dwt_lo(bufA, 8195, bufB);                // cA2, n=4101
    pool_hi_dwt(bufA, 8195, pout + 4 * 128); // cD2 -> level idx 4
    __syncthreads();
    dwt_lo(bufB, 4101, bufA);                // cA3, n=2054
    pool_hi_dwt(bufB, 4101, pout + 3 * 128); // cD3
    __syncthreads();
    dwt_lo(bufA, 2054, bufB);                // cA4, n=1030
    pool_hi_dwt(bufA, 2054, pout + 2 * 128); // cD4
    __syncthreads();
    dwt_lo(bufB, 1030, bufA);                // cA5, n=518
    pool_hi_dwt(bufB, 1030, pout + 1 * 128); // cD5
    __syncthreads();
    pool_plain(bufA, 518, pout + 0 * 128);   // cA5
}

// Kernel 2: per-level GEMM. C[b,o] = relu(A[b,:] . W[o,:] + bias[o])
// M=2048, N=128, K=128, f32 WMMA 16x16x4. Wave owns a 16x16 tile.
__global__ __launch_bounds__(256) void level_gemm_kernel(
    const float* __restrict__ pooled,   // [2048, 5, 128]
    const float* __restrict__ level_w,  // [5, 128, 128]  (o, p)
    const float* __restrict__ level_b,  // [5, 128]
    float* __restrict__ comb) {         // [2048, 640]
    const int l    = blockIdx.y;
    const int m0   = blockIdx.x * 16;
    const int wave = threadIdx.x >> 5;      // 8 waves -> 8 N-tiles of 16
    const int lane = threadIdx.x & 31;
    const int lr   = lane & 15;
    const int hi   = lane >> 4;             // K sub-pair select
    const int n    = wave * 16 + lr;

    const float* A    = pooled + (size_t)(m0 + lr) * 640 + (size_t)l * 128;
    const float* Bcol = level_w + ((size_t)l * 128 + n) * 128;  // B[k][n] = W[n][k]

    v8f c = {};
    for (int k = 0; k < 128; k += 4) {
        const int kb = k + hi * 2;
        v2f a, bf;
        a.x  = A[kb];    a.y  = A[kb + 1];
        bf.x = Bcol[kb]; bf.y = Bcol[kb + 1];
        c = __builtin_amdgcn_wmma_f32_16x16x4_f32(false, a, false, bf,
                                                  (short)0, c, false, false);
    }
    const float bias = level_b[l * 128 + n];
#pragma unroll
    for (int r = 0; r < 8; ++r) {
        const int row = m0 + r + hi * 8;
        float v = c[r] + bias;
        comb[(size_t)row * 640 + l * 128 + n] = v > 0.f ? v : 0.f;
    }
}

// Kernel 3: fusion GEMM. M=2048, N=512, K=640. out = relu(comb @ Wf^T + bf)
__global__ __launch_bounds__(256) void fusion_gemm_kernel(
    const float* __restrict__ comb,      // [2048, 640]
    const float* __restrict__ fusion_w,  // [512, 640]  (o, j)
    const float* __restrict__ fusion_b,  // [512]
    float* __restrict__ out) {           // [2048, 512]
    const int m0   = blockIdx.x * 16;
    const int wave = threadIdx.x >> 5;
    const int lane = threadIdx.x & 31;
    const int lr   = lane & 15;
    const int hi   = lane >> 4;
    const int n    = blockIdx.y * 128 + wave * 16 + lr;

    const float* A    = comb + (size_t)(m0 + lr) * 640;
    const float* Bcol = fusion_w + (size_t)n * 640;

    v8f c = {};
    for (int k = 0; k < 640; k += 4) {
        const int kb = k + hi * 2;
        v2f a, bf;
        a.x  = A[kb];    a.y  = A[kb + 1];
        bf.x = Bcol[kb]; bf.y = Bcol[kb + 1];
        c = __builtin_amdgcn_wmma_f32_16x16x4_f32(false, a, false, bf,
                                                  (short)0, c, false, false);
    }
    const float bias = fusion_b[n];
#pragma unroll
    for (int r = 0; r < 8; ++r) {
        const int row = m0 + r + hi * 8;
        float v = c[r] + bias;
        out[(size_t)row * 512 + n] = v > 0.f ? v : 0.f;
    }
}

extern "C" void kernel_launch(void* const* d_in, const int* in_sizes, int n_in,
                              void* d_out, int out_size, void* d_ws, size_t ws_size,
                              hipStream_t stream) {
    const float* x        = (const float*)d_in[0]; // [2048,16384]
    const float* level_w  = (const float*)d_in[1]; // [5,128,128]
    const float* level_b  = (const float*)d_in[2]; // [5,128]
    const float* fusion_w = (const float*)d_in[3]; // [512,640]
    const float* fusion_b = (const float*)d_in[4]; // [512]
    float* out = (float*)d_out;                    // [2048,512]

    float* pooled = (float*)d_ws;                  // [2048,5,128]
    float* comb   = pooled + (size_t)2048 * 640;   // [2048,640]

    dwt_pool_kernel<<<2048, 256, 0, stream>>>(x, pooled);
    level_gemm_kernel<<<dim3(2048 / 16, 5), 256, 0, stream>>>(pooled, level_w,
                                                              level_b, comb);
    fusion_gemm_kernel<<<dim3(2048 / 16, 512 / 128), 256, 0, stream>>>(
        comb, fusion_w, fusion_b, out);
}